// BayesRingRNN_15436112462538
// MI455X (gfx1250) — compile-verified
//
#include <hip/hip_runtime.h>
#include <math.h>

#define T_STEPS 1500
#define NX 80
#define BX 1024

typedef float v2f __attribute__((ext_vector_type(2)));
typedef float v8f __attribute__((ext_vector_type(8)));

// One wave (32 lanes) owns 16 batch rows and the full N=80 state.
// State r is held as five 16x16 f32 WMMA C/D-layout tiles:
//   lane 0-15: N = 16k + lane, rows M = vgpr (0..7)
//   lane 16-31: N = 16k + (lane-16), rows M = 8 + vgpr
__global__ __launch_bounds__(32) void ring_scan_kernel(
    const float* __restrict__ inp,   // (B, T, 2) float32: [:,:,0]=HD, [:,:,1]=AV
    const float* __restrict__ phi,   // (N) float32
    float* __restrict__ out,         // (B,T,N) followed by (B,N) final state
    float kz)
{
    const int  lane  = threadIdx.x;
    const int  nlane = lane & 15;
    const bool hi    = lane >= 16;
    const int  rbase = blockIdx.x * 16;   // 16 batch rows per wave
    const int  vb    = hi ? 8 : 0;        // row offset of this half's C/D slots

    __shared__ float sC[16], sS[16], sD[16];

    const float DEC0 = 1.0f - 0.01f / 3.0f;   // 1 - ALPHA*DT, ALPHA = 1/3
    const float K1   = 0.01f / 120.0f;        // (DT/(KP+KV)) * (2/N)
    const float KC   = 0.01f / 120.0f;        // DT * a_even * (2/N)
    const float KA   = 1.0f / 60.0f;          // a_odd * (2/N)

    float cph[5], sph[5];
    v8f   r[5];
    v2f   bmat[5];
#pragma unroll
    for (int k = 0; k < 5; ++k) {
        float p  = phi[16 * k + nlane];
        float cp = cosf(p), sp = sinf(p);
        cph[k] = cp;
        sph[k] = sp;
        float r0 = 10.0f * cp;                // KAPPA_0 * cos(phi)
#pragma unroll
        for (int v = 0; v < 8; ++v) r[k][v] = r0;
        // B operand (4x16, K x N): low lanes hold K=0 (VGPR0) / K=1 (VGPR1),
        // high lanes hold K=2 / K=3 which are zero.
        bmat[k].x = hi ? 0.0f : cp;
        bmat[k].y = hi ? 0.0f : sp;
    }

    // Per-slot output row pointers; per-step offset handled by t*NX + imm(16k).
    float* rowp[8];
#pragma unroll
    for (int v = 0; v < 8; ++v)
        rowp[v] = out + (size_t)(rbase + vb + v) * T_STEPS * NX + nlane;

    const float* ip = inp + (size_t)(rbase + nlane) * T_STEPS * 2;  // used by lanes < 16

    for (int t = 0; t < T_STEPS; ++t) {
        // ---- per-row projections c = r.cos(phi), s = r.sin(phi) ----
        float pc[8], ps[8];
#pragma unroll
        for (int v = 0; v < 8; ++v) { pc[v] = 0.0f; ps[v] = 0.0f; }
#pragma unroll
        for (int k = 0; k < 5; ++k)
#pragma unroll
            for (int v = 0; v < 8; ++v) {
                pc[v] = fmaf(r[k][v], cph[k], pc[v]);
                ps[v] = fmaf(r[k][v], sph[k], ps[v]);
            }
        // reduce across the 16-lane halves (masks 1,2,4,8 stay within a half)
#pragma unroll
        for (int m = 1; m < 16; m <<= 1)
#pragma unroll
            for (int v = 0; v < 8; ++v) {
                pc[v] += __shfl_xor(pc[v], m, 32);
                ps[v] += __shfl_xor(ps[v], m, 32);
            }

        // stage row sums: slot-indexed -> row-indexed via wave-private LDS
        if (lane == 0 || lane == 16) {
#pragma unroll
            for (int v = 0; v < 8; ++v) { sC[vb + v] = pc[v]; sS[vb + v] = ps[v]; }
        }
        __syncthreads();   // single-wave workgroup: lowers to S_NOP + waitcnt fencing

        // ---- per-row scalars (lane m handles batch row rbase+m) ----
        v2f amat; amat.x = 0.0f; amat.y = 0.0f;   // high lanes: A-operand K=2,3 == 0
        if (!hi) {
            float c  = sC[lane];
            float s  = sS[lane];
            float hd = ip[2 * t + 0];
            float av = ip[2 * t + 1];
            float nrm = __builtin_sqrtf(fmaf(c, c, s * s));
            sD[lane]  = DEC0 - K1 * nrm;
            float sh, ch;
            __sincosf(hd, &sh, &ch);
            amat.x = fmaf(kz, ch, KC * c - KA * av * s);
            amat.y = fmaf(kz, sh, KC * s + KA * av * c);
        }
        __syncthreads();

        float dv[8];
#pragma unroll
        for (int v = 0; v < 8; ++v) dv[v] = sD[vb + v];

        // ---- rank-2 update via WMMA: r = A(16x4) x B(4x16) + decay.*r ----
        size_t toff = (size_t)t * NX;
#pragma unroll
        for (int k = 0; k < 5; ++k) {
            v8f cacc;
#pragma unroll
            for (int v = 0; v < 8; ++v) cacc[v] = dv[v] * r[k][v];
            r[k] = __builtin_amdgcn_wmma_f32_16x16x4_f32(
                false, amat, false, bmat[k], (short)0, cacc, false, false);
        }

        // ---- stream r_new to out (B,T,N) ----
#pragma unroll
        for (int v = 0; v < 8; ++v)
#pragma unroll
            for (int k = 0; k < 5; ++k)
                rowp[v][toff + 16 * k] = r[k][v];
    }

    // ---- final state (1,B,N) appended after (B,T,N) ----
    float* o2 = out + (size_t)BX * T_STEPS * NX;
#pragma unroll
    for (int v = 0; v < 8; ++v)
#pragma unroll
        for (int k = 0; k < 5; ++k)
            o2[(size_t)(rbase + vb + v) * NX + 16 * k + nlane] = r[k][v];
}

// ---- host: reproduce reference's kappa_z = xi_inv(15*dt) bisection ----
static double xi_f(double a, double target) {
    double x = (a * 0.5) * (a * 0.5);
    double t0 = 1.0, t1 = a * 0.5, i0 = t0, i1 = t1;
    for (int k = 1; k < 30; ++k) {
        t0 *= x / (double)(k * k);
        t1 *= x / (double)(k * (k + 1));
        i0 += t0;
        i1 += t1;
    }
    return a * i1 / i0 - target;
}
static double xi_inv(double target) {
    double lo = 1e-3, hi = 50.0;
    for (int i = 0; i < 200; ++i) {
        double mid = 0.5 * (lo + hi);
        if (xi_f(lo, target) * xi_f(mid, target) <= 0.0) hi = mid;
        else lo = mid;
    }
    return 0.5 * (lo + hi);
}

extern "C" void kernel_launch(void* const* d_in, const int* in_sizes, int n_in,
                              void* d_out, int out_size, void* d_ws, size_t ws_size,
                              hipStream_t stream) {
    (void)in_sizes; (void)n_in; (void)out_size; (void)d_ws; (void)ws_size;
    const float* inp = (const float*)d_in[0];  // inputs (B,T,2)
    const float* phi = (const float*)d_in[4];  // phi (N)
    float* out = (float*)d_out;

    float kz = (float)xi_inv(15.0 * 0.01);     // ~0.5582

    dim3 grid(BX / 16), block(32);
    hipLaunchKernelGGL(ring_scan_kernel, grid, block, 0, stream, inp, phi, out, kz);
}